// TMAN2_10479720202425
// MI455X (gfx1250) — compile-verified
//
#include <hip/hip_runtime.h>
#include <math.h>
#include <stdint.h>

typedef float v2f __attribute__((ext_vector_type(2)));
typedef float v8f __attribute__((ext_vector_type(8)));

#define DEV __device__ __forceinline__

DEV v8f wmma4(v2f a, v2f b, v8f c) {
  // V_WMMA_F32_16X16X4_F32 : D(16x16 f32) = A(16x4) * B(4x16) + C
  return __builtin_amdgcn_wmma_f32_16x16x4_f32(
      /*neg_a=*/false, a, /*neg_b=*/false, b,
      /*c_mod=*/(short)0, c, /*reuse_a=*/false, /*reuse_b=*/false);
}

// ---- CDNA5 async global->LDS copy (ASYNCcnt-tracked) ------------------------
DEV void async_copy_b128(uint32_t lds_off, const void* gptr) {
  asm volatile("global_load_async_to_lds_b128 %0, %1, off"
               :: "v"(lds_off), "v"((uint64_t)(uintptr_t)gptr)
               : "memory");
}
DEV void async_copy_b32(uint32_t lds_off, const void* gptr) {
  asm volatile("global_load_async_to_lds_b32 %0, %1, off"
               :: "v"(lds_off), "v"((uint64_t)(uintptr_t)gptr)
               : "memory");
}
DEV void wait_async0() {
  asm volatile("s_wait_asynccnt 0x0" ::: "memory");
}

// acc += A(16xK row-major, stride lda) * B(Kx16 slice of row-major W, stride ldb, col base cb)
// K = 4*KSTEPS exact.  li = lane&15, kh = lane>>4.
template <int KSTEPS>
DEV void gemm_AB(v8f& acc, const float* __restrict__ As, int lda,
                 const float* __restrict__ Bg, int ldb, int cb, int li, int kh) {
#pragma unroll 4
  for (int ks = 0; ks < KSTEPS; ++ks) {
    const int k = ks * 4 + 2 * kh;
    v2f a, b;
    a.x = As[li * lda + k];
    a.y = As[li * lda + k + 1];
    b.x = Bg[k * ldb + cb + li];
    b.y = Bg[(k + 1) * ldb + cb + li];
    acc = wmma4(a, b, acc);
  }
}

// Same but K is arbitrary (zero-padded tail).
DEV void gemm_AB_bnd(v8f& acc, const float* __restrict__ As, int lda,
                     const float* __restrict__ Bg, int ldb, int cb,
                     int K, int li, int kh) {
  const int ksteps = (K + 3) >> 2;
  for (int ks = 0; ks < ksteps; ++ks) {
    const int k = ks * 4 + 2 * kh;
    v2f a, b;
    a.x = (k < K)     ? As[li * lda + k]             : 0.0f;
    a.y = (k + 1 < K) ? As[li * lda + k + 1]         : 0.0f;
    b.x = (k < K)     ? Bg[k * ldb + cb + li]        : 0.0f;
    b.y = (k + 1 < K) ? Bg[(k + 1) * ldb + cb + li]  : 0.0f;
    acc = wmma4(a, b, acc);
  }
}

DEV float sigf(float x) { return 1.0f / (1.0f + expf(-x)); }

struct Params {
  const float* x;
  const float *Wl, *bWl, *Ul, *bUl, *Vl, *bVl;
  const float *Wa, *bWa, *Ua, *bUa, *Va, *bVa;
  const float *Wv, *bWv, *Uv, *bUv, *Vv, *bVv;
  const float *nlW1, *nlb1, *nlW2, *nlb2;
  const float *naW1, *nab1, *naW2, *nab2;
  const float *nvW1, *nvb1, *nvW2, *nvb2;
  const float *attW1, *attb1, *attW2, *attb2;
  const float *outW1, *outb1, *outW2, *outb2;
  float* out;
};

#define T_STEPS 128
#define NBATCH  2048
#define DXTOT   409   // 300 + 74 + 35
#define XPAD    412   // row stride in LDS x tile (412*4 = 103 * 16B, aligned rows)
#define ROWS    16

__global__ __launch_bounds__(256) void tman_fused(Params p) {
  __shared__ __align__(16) float sh_x[2][ROWS][XPAD];  // double-buffered x tile
  __shared__ float sh_hl[ROWS][128];
  __shared__ float sh_ha[ROWS][64];
  __shared__ float sh_hv[ROWS][64];
  __shared__ float sh_z [ROWS][64];
  __shared__ float sh_cs[ROWS][256];   // c_l | c_a | c_v  (cols 0..127 | 128..191 | 192..255)
  __shared__ float sh_hid[ROWS][128];  // MLP hidden scratch (reused per stage)
  __shared__ float sh_sl[ROWS][64];
  __shared__ float sh_sa[ROWS][64];
  __shared__ float sh_sv[ROWS][64];
  __shared__ float sh_att[ROWS][4];

  const int tid  = threadIdx.x;
  const int wid  = tid >> 5;
  const int lane = tid & 31;
  const int li   = lane & 15;
  const int kh   = lane >> 4;
  const int rowbase = blockIdx.x * ROWS;

  // zero state
  for (int i = tid; i < ROWS * 128; i += 256) { sh_hl[0][i] = 0.f; sh_hid[0][i] = 0.f; }
  for (int i = tid; i < ROWS * 64; i += 256) {
    sh_ha[0][i] = 0.f; sh_hv[0][i] = 0.f; sh_z[0][i] = 0.f;
    sh_sl[0][i] = 0.f; sh_sa[0][i] = 0.f; sh_sv[0][i] = 0.f;
  }
  for (int i = tid; i < ROWS * 256; i += 256) sh_cs[0][i] = 0.f;

  // ---- per-thread constant setup -------------------------------------------
  const int colL = wid * 16 + li;
  const float bLf = p.bWl[colL]       + p.bUl[colL]       + p.bVl[colL];
  const float bLi = p.bWl[128 + colL] + p.bUl[128 + colL] + p.bVl[128 + colL];
  const float bLo = p.bWl[256 + colL] + p.bUl[256 + colL] + p.bVl[256 + colL];
  const float bLc = p.bWl[384 + colL] + p.bUl[384 + colL] + p.bVl[384 + colL];

  // Second modality: waves 0-3 -> A (DHA=64), waves 4-7 -> V (DHV=64)
  const bool isA  = (wid < 4);
  const int  q2   = isA ? wid : wid - 4;
  const int  col2 = q2 * 16 + li;
  const float* W2  = isA ? p.Wa  : p.Wv;
  const float* U2  = isA ? p.Ua  : p.Uv;
  const float* V2  = isA ? p.Va  : p.Vv;
  const float* bW2 = isA ? p.bWa : p.bWv;
  const float* bU2 = isA ? p.bUa : p.bUv;
  const float* bV2 = isA ? p.bVa : p.bVv;
  float* h2 = isA ? &sh_ha[0][0] : &sh_hv[0][0];
  const int K2     = isA ? 74  : 35;
  const int xoff2  = isA ? 300 : 374;
  const int csoff2 = isA ? 128 : 192;
  const float b2f = bW2[col2]       + bU2[col2]       + bV2[col2];
  const float b2i = bW2[64 + col2]  + bU2[64 + col2]  + bV2[64 + col2];
  const float b2o = bW2[128 + col2] + bU2[128 + col2] + bV2[128 + col2];
  const float b2c = bW2[192 + col2] + bU2[192 + col2] + bV2[192 + col2];

  // ---- stage x tile for t=0 asynchronously into LDS buffer 0 ---------------
  {
    const float* src = p.x + (size_t)rowbase * DXTOT;
    float* dst = &sh_x[0][0][0];
    for (int i = tid; i < ROWS * 102; i += 256) {
      const int row = i / 102, ch = i - row * 102;
      async_copy_b128((uint32_t)(uintptr_t)(dst + row * XPAD + ch * 4),
                      src + (size_t)row * DXTOT + ch * 4);
    }
    if (tid < ROWS)
      async_copy_b32((uint32_t)(uintptr_t)(dst + tid * XPAD + 408),
                     src + (size_t)tid * DXTOT + 408);
  }

  // ---- recurrent scan ------------------------------------------------------
  for (int t = 0; t < T_STEPS; ++t) {
    wait_async0();      // this wave's async copies into buffer (t&1) done
    __syncthreads();    // -> whole tile visible to all waves

    const float* xs = &sh_x[t & 1][0][0];

    // kick off async staging of next timestep's x tile (hidden behind compute)
    if (t + 1 < T_STEPS) {
      const float* src = p.x + ((size_t)(t + 1) * NBATCH + rowbase) * DXTOT;
      float* dst = &sh_x[(t + 1) & 1][0][0];
      for (int i = tid; i < ROWS * 102; i += 256) {
        const int row = i / 102, ch = i - row * 102;
        async_copy_b128((uint32_t)(uintptr_t)(dst + row * XPAD + ch * 4),
                        src + (size_t)row * DXTOT + ch * 4);
      }
      if (tid < ROWS)
        async_copy_b32((uint32_t)(uintptr_t)(dst + tid * XPAD + 408),
                       src + (size_t)tid * DXTOT + 408);
    }

    // -- gate pre-activations: s = x@W + h@U + z@V (biases folded in later) --
    v8f aL[4] = {{0.f}, {0.f}, {0.f}, {0.f}};
#pragma unroll
    for (int g = 0; g < 4; ++g) {
      const int cb = g * 128 + wid * 16;
      gemm_AB_bnd(aL[g], xs, XPAD, p.Wl, 512, cb, 300, li, kh);
      gemm_AB<32>(aL[g], &sh_hl[0][0], 128, p.Ul, 512, cb, li, kh);
      gemm_AB<16>(aL[g], &sh_z[0][0],   64, p.Vl, 512, cb, li, kh);
    }
    v8f aS[4] = {{0.f}, {0.f}, {0.f}, {0.f}};
#pragma unroll
    for (int g = 0; g < 4; ++g) {
      const int cb = g * 64 + q2 * 16;
      gemm_AB_bnd(aS[g], xs + xoff2, XPAD, W2, 256, cb, K2, li, kh);
      gemm_AB<16>(aS[g], h2,           64, U2, 256, cb, li, kh);
      gemm_AB<16>(aS[g], &sh_z[0][0],  64, V2, 256, cb, li, kh);
    }
    __syncthreads();  // all reads of h_l/h_a/h_v/z complete

    // -- gates + cell/hidden update (register-resident, no s staging) --------
#pragma unroll
    for (int r = 0; r < 8; ++r) {
      const int row = r + 8 * kh;
      {
        const float f  = sigf(aL[0][r] + bLf);
        const float i  = sigf(aL[1][r] + bLi);
        const float o  = sigf(aL[2][r] + bLo);
        const float ch = tanhf(aL[3][r] + bLc);
        const float c  = f * sh_cs[row][colL] + i * ch;
        sh_cs[row][colL] = c;
        sh_hl[row][colL] = tanhf(c) * o;
      }
      {
        const float f  = sigf(aS[0][r] + b2f);
        const float i  = sigf(aS[1][r] + b2i);
        const float o  = sigf(aS[2][r] + b2o);
        const float ch = tanhf(aS[3][r] + b2c);
        const float c  = f * sh_cs[row][csoff2 + col2] + i * ch;
        sh_cs[row][csoff2 + col2] = c;
        h2[row * 64 + col2] = tanhf(c) * o;
      }
    }
    __syncthreads();

    // -- nl MLP: s_l = relu(c_l @ nlW1 + b) @ nlW2 + b -----------------------
    {
      v8f a = {0.f};
      const int cb = wid * 16;
      gemm_AB<32>(a, &sh_cs[0][0], 256, p.nlW1, 128, cb, li, kh);
#pragma unroll
      for (int r = 0; r < 8; ++r) {
        const float v = a[r] + p.nlb1[cb + li];
        sh_hid[r + 8 * kh][cb + li] = v > 0.f ? v : 0.f;
      }
    }
    __syncthreads();
    if (wid < 4) {
      v8f a = {0.f};
      const int cb = wid * 16;
      gemm_AB<32>(a, &sh_hid[0][0], 128, p.nlW2, 64, cb, li, kh);
#pragma unroll
      for (int r = 0; r < 8; ++r) sh_sl[r + 8 * kh][cb + li] = a[r] + p.nlb2[cb + li];
    }
    __syncthreads();

    // -- na MLP --------------------------------------------------------------
    {
      v8f a = {0.f};
      const int cb = wid * 16;
      gemm_AB<16>(a, &sh_cs[0][128], 256, p.naW1, 128, cb, li, kh);
#pragma unroll
      for (int r = 0; r < 8; ++r) {
        const float v = a[r] + p.nab1[cb + li];
        sh_hid[r + 8 * kh][cb + li] = v > 0.f ? v : 0.f;
      }
    }
    __syncthreads();
    if (wid < 4) {
      v8f a = {0.f};
      const int cb = wid * 16;
      gemm_AB<32>(a, &sh_hid[0][0], 128, p.naW2, 64, cb, li, kh);
#pragma unroll
      for (int r = 0; r < 8; ++r) sh_sa[r + 8 * kh][cb + li] = a[r] + p.nab2[cb + li];
    }
    __syncthreads();

    // -- nv MLP --------------------------------------------------------------
    {
      v8f a = {0.f};
      const int cb = wid * 16;
      gemm_AB<16>(a, &sh_cs[0][192], 256, p.nvW1, 128, cb, li, kh);
#pragma unroll
      for (int r = 0; r < 8; ++r) {
        const float v = a[r] + p.nvb1[cb + li];
        sh_hid[r + 8 * kh][cb + li] = v > 0.f ? v : 0.f;
      }
    }
    __syncthreads();
    if (wid < 4) {
      v8f a = {0.f};
      const int cb = wid * 16;
      gemm_AB<32>(a, &sh_hid[0][0], 128, p.nvW2, 64, cb, li, kh);
#pragma unroll
      for (int r = 0; r < 8; ++r) sh_sv[r + 8 * kh][cb + li] = a[r] + p.nvb2[cb + li];
    }
    __syncthreads();

    // -- attention MLP: logits = relu(cs @ attW1 + b) @ attW2 + b ------------
    {
      v8f a = {0.f};
      const int cb = wid * 16;
      gemm_AB<64>(a, &sh_cs[0][0], 256, p.attW1, 128, cb, li, kh);
#pragma unroll
      for (int r = 0; r < 8; ++r) {
        const float v = a[r] + p.attb1[cb + li];
        sh_hid[r + 8 * kh][cb + li] = v > 0.f ? v : 0.f;
      }
    }
    __syncthreads();
    if (wid == 0) {  // 128 x 3 tail, B cols masked to 3
      v8f a = {0.f};
#pragma unroll 4
      for (int ks = 0; ks < 32; ++ks) {
        const int k = ks * 4 + 2 * kh;
        v2f av, bv;
        av.x = sh_hid[li][k];
        av.y = sh_hid[li][k + 1];
        bv.x = (li < 3) ? p.attW2[k * 3 + li] : 0.f;
        bv.y = (li < 3) ? p.attW2[(k + 1) * 3 + li] : 0.f;
        a = wmma4(av, bv, a);
      }
#pragma unroll
      for (int r = 0; r < 8; ++r)
        if (li < 3) sh_att[r + 8 * kh][li] = a[r] + p.attb2[li];
    }
    __syncthreads();

    // -- softmax over 3, then senti -> z -------------------------------------
    if (tid < ROWS) {
      float e0 = sh_att[tid][0], e1 = sh_att[tid][1], e2 = sh_att[tid][2];
      const float mx = fmaxf(e0, fmaxf(e1, e2));
      e0 = expf(e0 - mx); e1 = expf(e1 - mx); e2 = expf(e2 - mx);
      const float inv = 1.0f / (e0 + e1 + e2);
      sh_att[tid][0] = e0 * inv; sh_att[tid][1] = e1 * inv; sh_att[tid][2] = e2 * inv;
    }
    __syncthreads();
    for (int i = tid; i < ROWS * 64; i += 256) {
      const int row = i >> 6, k = i & 63;
      sh_z[row][k] = sh_att[row][0] * sh_sl[row][k] +
                     sh_att[row][1] * sh_sa[row][k] +
                     sh_att[row][2] * sh_sv[row][k];
    }
    __syncthreads();
  }

  // ---- output head: relu([h_l|h_a|h_v|z] @ outW1 + b) @ outW2 + b ----------
  for (int i = tid; i < ROWS * 128; i += 256) {
    const int row = i >> 7, c = i & 127;
    float s = p.outb1[c];
    for (int k = 0; k < 128; ++k) s += sh_hl[row][k] * p.outW1[k * 128 + c];
    for (int k = 0; k < 64;  ++k) s += sh_ha[row][k] * p.outW1[(128 + k) * 128 + c];
    for (int k = 0; k < 64;  ++k) s += sh_hv[row][k] * p.outW1[(192 + k) * 128 + c];
    for (int k = 0; k < 64;  ++k) s += sh_z[row][k]  * p.outW1[(256 + k) * 128 + c];
    sh_hid[row][c] = s > 0.f ? s : 0.f;
  }
  __syncthreads();
  if (tid < ROWS) {
    float s = p.outb2[0];
    for (int k = 0; k < 128; ++k) s += sh_hid[tid][k] * p.outW2[k];
    p.out[rowbase + tid] = s;
  }
}

extern "C" void kernel_launch(void* const* d_in, const int* in_sizes, int n_in,
                              void* d_out, int out_size, void* d_ws, size_t ws_size,
                              hipStream_t stream) {
  (void)in_sizes; (void)n_in; (void)d_ws; (void)ws_size; (void)out_size;
  Params p;
  const float** pp = reinterpret_cast<const float**>(&p);
  for (int i = 0; i < 39; ++i) pp[i] = (const float*)d_in[i];
  p.out = (float*)d_out;

  tman_fused<<<dim3(NBATCH / ROWS), dim3(256), 0, stream>>>(p);
}